// GraphGRUCell_37297495998610
// MI455X (gfx1250) — compile-verified
//
#include <hip/hip_runtime.h>
#include <math.h>

#define HD 96   // feature width (d == h == 96)

typedef float v2f __attribute__((ext_vector_type(2)));
typedef float v8f __attribute__((ext_vector_type(8)));

// ---------------------------------------------------------------------------
// Degree / normalization:  deg = 1 (self loop) + count(dst);  dinv = rsqrt(deg)
// ---------------------------------------------------------------------------
__global__ void k_deg_init(float* __restrict__ dinv, int n) {
    int i = blockIdx.x * blockDim.x + threadIdx.x;
    if (i < n) dinv[i] = 1.0f;                       // self-loop contribution
}

__global__ void k_deg_count(const int* __restrict__ dst, float* __restrict__ dinv, int E) {
    int e = blockIdx.x * blockDim.x + threadIdx.x;
    if (e < E) atomicAdd(&dinv[dst[e]], 1.0f);
}

__global__ void k_rsqrt(float* __restrict__ dinv, int n) {
    int i = blockIdx.x * blockDim.x + threadIdx.x;
    if (i < n) dinv[i] = rsqrtf(dinv[i]);            // deg >= 1 always
}

// ---------------------------------------------------------------------------
// WMMA fp32 GEMM:  OUT[n x 96] = A[n x K] @ W[K x 96],  K = 96 or 192 (concat)
// One 192-thread block = 6 waves; wave w computes the 16x16 tile at
// (blockIdx.x*16, w*16) via V_WMMA_F32_16X16X4_F32, K stepped by 4.
// A layout (ISA 7.12.2, 32-bit 16x4): lanes 0-15 carry K=k0,k0+1 for row
// M=lane&15; lanes 16-31 carry K=k0+2,k0+3.  B (4x16) mirrors the K split.
// Straight-line, fully unrolled segments: no branches in the hot loop.
// ---------------------------------------------------------------------------
__device__ __forceinline__ v8f wmma_f32(v2f a, v2f b, v8f c) {
    return __builtin_amdgcn_wmma_f32_16x16x4_f32(
        /*neg_a=*/false, a, /*neg_b=*/false, b,
        /*c_mod=*/(short)0, c, /*reuse_a=*/false, /*reuse_b=*/false);
}

// Abase = &A[row*96 + koff]  (8B-aligned: koff in {0,2}, row*96 even)
// Bbase = &W[koff*96 + col]
__device__ __forceinline__ void gemm_seg96(const float* __restrict__ Abase,
                                           const float* __restrict__ Bbase,
                                           v8f& acc) {
#pragma unroll
    for (int k0 = 0; k0 < HD; k0 += 4) {
        const v2f a = *(const v2f*)(Abase + k0);           // A[row][ka..ka+1]
        v2f b;
        b.x = Bbase[k0 * HD];                              // W[ka  ][col]
        b.y = Bbase[k0 * HD + HD];                         // W[ka+1][col]
        acc = wmma_f32(a, b, acc);
    }
}

template <bool CONCAT>
__global__ void __launch_bounds__(192)
k_gemm_wmma(const float* __restrict__ A0, const float* __restrict__ A1,
            const float* __restrict__ W, float* __restrict__ out, int n)
{
    const int lane  = threadIdx.x & 31;
    const int wave  = threadIdx.x >> 5;          // 0..5 -> N tile
    const int l15   = lane & 15;
    const int koff  = (lane < 16) ? 0 : 2;       // K sub-offset per half-wave
    const int mbase = blockIdx.x * 16;
    const int nbase = wave * 16;
    const int col   = nbase + l15;

    int row = mbase + l15;
    if (row >= n) row = n - 1;                   // clamp (keeps EXEC all-ones)

    v8f acc = {};
    gemm_seg96(A0 + (size_t)row * HD + koff, W + (size_t)koff * HD + col, acc);
    if (CONCAT)
        gemm_seg96(A1 + (size_t)row * HD + koff,
                   W + (size_t)(HD + koff) * HD + col, acc);

    // C/D layout: VGPR r -> M = mbase + r (+8 for lanes 16-31), N = col
    const int rbase = mbase + ((lane < 16) ? 0 : 8);
#pragma unroll
    for (int r = 0; r < 8; ++r) {
        const int rr = rbase + r;
        if (rr < n) out[(size_t)rr * HD + col] = acc[r];
    }
}

// ---------------------------------------------------------------------------
// Propagation: agg[i] = H[i]*dinv[i]^2 (self loop), then edge scatter-add.
// ---------------------------------------------------------------------------
__global__ void k_selfloop(const float* __restrict__ H, const float* __restrict__ dinv,
                           float* __restrict__ agg, int cnt) {
    int idx = blockIdx.x * blockDim.x + threadIdx.x;
    if (idx < cnt) {
        const float di = dinv[idx / HD];
        agg[idx] = H[idx] * di * di;
    }
}

// one wave per edge; lane covers columns {lane, lane+32, lane+64}
__global__ void k_scatter(const float* __restrict__ H,
                          const int* __restrict__ src, const int* __restrict__ dst,
                          const float* __restrict__ dinv, float* __restrict__ agg, int E)
{
    const int gid  = blockIdx.x * blockDim.x + threadIdx.x;
    const int e    = gid >> 5;
    const int lane = gid & 31;
    if (e >= E) return;
    const int s = src[e];
    const int d = dst[e];
    const float nrm = dinv[s] * dinv[d];
    const float* hp = H   + (size_t)s * HD;
    float*       op = agg + (size_t)d * HD;
    atomicAdd(&op[lane],      hp[lane]      * nrm);
    atomicAdd(&op[lane + 32], hp[lane + 32] * nrm);
    atomicAdd(&op[lane + 64], hp[lane + 64] * nrm);
}

// ---------------------------------------------------------------------------
// Elementwise kernels
// ---------------------------------------------------------------------------
__global__ void k_bias(const float* __restrict__ agg, const float* __restrict__ b,
                       float* __restrict__ out, int cnt) {
    int idx = blockIdx.x * blockDim.x + threadIdx.x;
    if (idx < cnt) out[idx] = agg[idx] + b[idx % HD];
}

__global__ void k_sigmoid_bias(const float* __restrict__ agg, const float* __restrict__ b,
                               float* __restrict__ out, int cnt) {
    int idx = blockIdx.x * blockDim.x + threadIdx.x;
    if (idx < cnt) {
        const float z = agg[idx] + b[idx % HD];
        out[idx] = 1.0f / (1.0f + expf(-z));
    }
}

__global__ void k_mul(const float* __restrict__ a, const float* __restrict__ b,
                      float* __restrict__ out, int cnt) {
    int idx = blockIdx.x * blockDim.x + threadIdx.x;
    if (idx < cnt) out[idx] = a[idx] * b[idx];
}

__global__ void k_final(const float* __restrict__ g, const float* __restrict__ hprev,
                        const float* __restrict__ agg, const float* __restrict__ b,
                        float* __restrict__ out, int cnt) {
    int idx = blockIdx.x * blockDim.x + threadIdx.x;
    if (idx < cnt) {
        const float u = g[idx];
        const float c = tanhf(agg[idx] + b[idx % HD]);
        out[idx] = u * hprev[idx] + (1.0f - u) * c;
    }
}

// ---------------------------------------------------------------------------
// Host-side orchestration (all kernels on `stream`; scratch in d_ws)
// ---------------------------------------------------------------------------
extern "C" void kernel_launch(void* const* d_in, const int* in_sizes, int n_in,
                              void* d_out, int out_size, void* d_ws, size_t ws_size,
                              hipStream_t stream) {
    const float* x      = (const float*)d_in[0];
    const int*   ei     = (const int*)  d_in[1];
    const float* h_prev = (const float*)d_in[2];
    const float* Wx     = (const float*)d_in[3];
    const float* bx     = (const float*)d_in[4];
    const float* Wuh    = (const float*)d_in[5];
    const float* buh    = (const float*)d_in[6];
    const float* Wch    = (const float*)d_in[7];
    const float* bch    = (const float*)d_in[8];
    float*       out    = (float*)d_out;

    const int n = in_sizes[0] / HD;
    const int E = in_sizes[1] / 2;
    const int* src = ei;
    const int* dst = ei + E;

    float* ws   = (float*)d_ws;
    float* dinv = ws;                         // [n]
    float* xg   = dinv + n;                   // [n*96] x_gcn (persistent)
    float* g    = xg + (size_t)n * HD;        // [n*96] gate u==r (persistent)
    float* Hm   = g  + (size_t)n * HD;        // [n*96] GEMM output / message src
    float* agg  = Hm + (size_t)n * HD;        // [n*96] scatter accumulator / rh

    const int cnt        = n * HD;
    const int nb_n       = (n + 255) / 256;
    const int nb_E       = (E + 255) / 256;
    const int nb_cnt     = (cnt + 255) / 256;
    const int nb_scatter = (E * 32 + 255) / 256;    // wave per edge
    const int mtiles     = (n + 15) / 16;

    // normalization coefficients
    k_deg_init <<<nb_n, 256, 0, stream>>>(dinv, n);
    k_deg_count<<<nb_E, 256, 0, stream>>>(dst, dinv, E);
    k_rsqrt    <<<nb_n, 256, 0, stream>>>(dinv, n);

    // ---- GCN 1: x_gcn = prop(x @ Wx) + bx
    k_gemm_wmma<false><<<mtiles, 192, 0, stream>>>(x, nullptr, Wx, Hm, n);
    k_selfloop <<<nb_cnt, 256, 0, stream>>>(Hm, dinv, agg, cnt);
    k_scatter  <<<nb_scatter, 256, 0, stream>>>(Hm, src, dst, dinv, agg, E);
    k_bias     <<<nb_cnt, 256, 0, stream>>>(agg, bx, xg, cnt);

    // ---- GCN 2: g = sigmoid(prop([x_gcn, h_prev] @ Wuh) + buh)   (u == r)
    k_gemm_wmma<true><<<mtiles, 192, 0, stream>>>(xg, h_prev, Wuh, Hm, n);
    k_selfloop <<<nb_cnt, 256, 0, stream>>>(Hm, dinv, agg, cnt);
    k_scatter  <<<nb_scatter, 256, 0, stream>>>(Hm, src, dst, dinv, agg, E);
    k_sigmoid_bias<<<nb_cnt, 256, 0, stream>>>(agg, buh, g, cnt);

    // ---- GCN 3: c = tanh(prop([x_gcn, g .* h_prev] @ Wch) + bch)
    //      out = g*h_prev + (1-g)*c
    k_mul      <<<nb_cnt, 256, 0, stream>>>(g, h_prev, agg, cnt);   // rh in agg
    k_gemm_wmma<true><<<mtiles, 192, 0, stream>>>(xg, agg, Wch, Hm, n);
    k_selfloop <<<nb_cnt, 256, 0, stream>>>(Hm, dinv, agg, cnt);    // overwrites rh
    k_scatter  <<<nb_scatter, 256, 0, stream>>>(Hm, src, dst, dinv, agg, E);
    k_final    <<<nb_cnt, 256, 0, stream>>>(g, h_prev, agg, bch, out, cnt);
}